// DecodePredictions_66228395704543
// MI455X (gfx1250) — compile-verified
//
#include <hip/hip_runtime.h>
#include <math.h>

// ---------------- constants (match reference) ----------------
#define K_PER    200           // max_detections_per_class
#define MAX_DET  200           // max_detections
#define KPAD     208           // 13 * 16  (pad K_PER to WMMA tiles)
#define NTILE    13
#define ROWW     7             // 7 * 32 = 224 bits >= KPAD
#define CAND_MAX 1024
#define NBUCKET  2048
#define CONF_T   0.05f
#define IOU_T    0.5f

typedef float v2f __attribute__((ext_vector_type(2)));
typedef float v8f __attribute__((ext_vector_type(8)));

// monotonic order-preserving float->uint key
__device__ __forceinline__ unsigned flipKey(float x) {
    unsigned u = __float_as_uint(x);
    return u ^ ((u >> 31) ? 0xFFFFFFFFu : 0x80000000u);
}
__device__ __forceinline__ float unflipKey(unsigned k) {
    unsigned u = (k & 0x80000000u) ? (k ^ 0x80000000u) : ~k;
    return __uint_as_float(u);
}

// ascending bitonic sort of CAND_MAX (key,val) pairs in LDS.
// keys are stored as ~flipKey so ascending == descending by original value.
__device__ __forceinline__ void bitonic_sort_pairs(unsigned* key, int* val,
                                                   int tid, int nthr) {
    for (unsigned k = 2; k <= CAND_MAX; k <<= 1) {
        for (unsigned j = k >> 1; j > 0; j >>= 1) {
            __syncthreads();
            for (unsigned i = (unsigned)tid; i < CAND_MAX; i += nthr) {
                unsigned ixj = i ^ j;
                if (ixj > i) {
                    bool up = ((i & k) == 0);
                    unsigned ki = key[i], kj = key[ixj];
                    bool sw = up ? (ki > kj) : (ki < kj);
                    if (sw) {
                        key[i] = kj; key[ixj] = ki;
                        int t = val[i]; val[i] = val[ixj]; val[ixj] = t;
                    }
                }
            }
        }
    }
    __syncthreads();
}

// ---------------------------------------------------------------------------
// Kernel 1: per-(batch,class) top-200 of A logits.
// Histogram (2048 buckets on flipped float bits) -> pivot -> collect -> sort.
// One block per (b,c); 2 streaming passes over the class column. 196 MB of
// logits ~= the 192 MB L2, and the 640 blocks walk it in near-lockstep, so
// pass 2 mostly hits L2.
// ---------------------------------------------------------------------------
__global__ void __launch_bounds__(256)
topk_per_class(const float* __restrict__ logits, int A, int C,
               float* __restrict__ ws_vals,   // [B*C][256] raw logits, sorted desc
               int*   __restrict__ ws_idx) {  // [B*C][256] anchor indices
    const int bc  = blockIdx.x;
    const int b   = bc / C;
    const int c   = bc % C;
    const int tid = threadIdx.x;

    __shared__ unsigned hist[NBUCKET];
    __shared__ unsigned candKey[CAND_MAX];
    __shared__ int      candIdx[CAND_MAX];
    __shared__ int      candCnt;
    __shared__ unsigned pivotKey;

    for (int i = tid; i < NBUCKET; i += blockDim.x) hist[i] = 0u;
    if (tid == 0) candCnt = 0;
    __syncthreads();

    const float* base = logits + (size_t)b * A * C + c;

    // pass 1: histogram (bandwidth-critical; prefetch ahead — speculative,
    // OOB addresses silently dropped per CDNA5 ISA 10.5)
    for (int a = tid; a < A; a += blockDim.x) {
        __builtin_prefetch(base + (size_t)(a + 2048) * C, 0, 1);
        unsigned key = flipKey(base[(size_t)a * C]);
        atomicAdd(&hist[key >> 21], 1u);
    }
    __syncthreads();

    if (tid == 0) {
        unsigned cum = 0; int p = 0;
        for (int i = NBUCKET - 1; i >= 0; --i) {
            cum += hist[i];
            if (cum >= K_PER) { p = i; break; }
        }
        pivotKey = ((unsigned)p) << 21;
    }
    __syncthreads();
    const unsigned pk = pivotKey;

    // pass 2: collect candidates (>= pivot bucket start)
    for (int a = tid; a < A; a += blockDim.x) {
        unsigned key = flipKey(base[(size_t)a * C]);
        if (key >= pk) {
            int pos = atomicAdd(&candCnt, 1);
            if (pos < CAND_MAX) { candKey[pos] = ~key; candIdx[pos] = a; }
        }
    }
    __syncthreads();
    int cnt = candCnt; if (cnt > CAND_MAX) cnt = CAND_MAX;
    for (int i = cnt + tid; i < CAND_MAX; i += blockDim.x) {
        candKey[i] = 0xFFFFFFFFu; candIdx[i] = 0;
    }
    __syncthreads();

    bitonic_sort_pairs(candKey, candIdx, tid, blockDim.x);

    for (int k = tid; k < 256; k += blockDim.x) {
        float v = -3.0e38f; int a = 0;
        if (k < K_PER && k < cnt) { v = unflipKey(~candKey[k]); a = candIdx[k]; }
        ws_vals[(size_t)bc * 256 + k] = v;
        ws_idx [(size_t)bc * 256 + k] = a;
    }
}

// ---------------------------------------------------------------------------
// Kernel 2: per-(batch,class) box decode + greedy NMS over the top-200.
// union[i][j] = area_i + area_j - inter : the outer-sum comes from
// V_WMMA_F32_16X16X4_F32 (A = [area_i | 1 | 0 | 0], B = [1 ; area_j ; 0 ; 0]).
// IoU test is division-free: inter/u > T  <=>  inter > T*u  (u > 0).
// Suppression bits gathered per half-wave via ballot -> 2 ds_or per D-row.
// ---------------------------------------------------------------------------
__global__ void __launch_bounds__(256)
nms_per_class(const float* __restrict__ reg,      // [B*A*4]
              const float* __restrict__ anchors,  // [A*4] (cx,cy,w,h)
              const float* __restrict__ ws_vals,
              const int*   __restrict__ ws_idx,
              int A, int C,
              float* __restrict__ ws_scores,      // [B*C*200]
              float* __restrict__ ws_boxes) {     // [B*C*200*4]
    const int bc  = blockIdx.x;
    const int b   = bc / C;
    const int tid = threadIdx.x;

    __shared__ float sx1[KPAD], sy1[KPAD], sx2[KPAD], sy2[KPAD];
    __shared__ float sarea[KPAD], sscore[KPAD];
    __shared__ unsigned rows[KPAD][ROWW];
    __shared__ unsigned keepw[ROWW];

    unsigned* rp = &rows[0][0];
    for (int i = tid; i < KPAD * ROWW; i += blockDim.x) rp[i] = 0u;

    // decode boxes for the selected anchors
    for (int k = tid; k < KPAD; k += blockDim.x) {
        float x1 = 0.f, y1 = 0.f, x2 = 0.f, y2 = 0.f, score = -1.f;
        if (k < K_PER) {
            float lg = ws_vals[(size_t)bc * 256 + k];
            int   a  = ws_idx [(size_t)bc * 256 + k];
            score = 1.f / (1.f + __expf(-lg));
            float acx = anchors[a * 4 + 0], acy = anchors[a * 4 + 1];
            float aw  = anchors[a * 4 + 2], ah  = anchors[a * 4 + 3];
            const float* r = reg + ((size_t)b * A + a) * 4;
            float cx = r[0] * 0.1f * aw + acx;
            float cy = r[1] * 0.1f * ah + acy;
            float w  = __expf(r[2] * 0.2f) * aw;
            float h  = __expf(r[3] * 0.2f) * ah;
            x1 = cx - 0.5f * w; y1 = cy - 0.5f * h;
            x2 = cx + 0.5f * w; y2 = cy + 0.5f * h;
            float* ob = ws_boxes + ((size_t)bc * K_PER + k) * 4;
            ob[0] = x1; ob[1] = y1; ob[2] = x2; ob[3] = y2;
        }
        sx1[k] = x1; sy1[k] = y1; sx2[k] = x2; sy2[k] = y2;
        sarea[k] = (x2 - x1) * (y2 - y1);
        sscore[k] = score;
    }
    __syncthreads();

    // suppression matrix: 13x13 tiles of 16x16, one WMMA per tile.
    const int wave = tid >> 5, lane = tid & 31;
    const int half = lane >> 4, l16 = lane & 15;
    const int nwaves = blockDim.x >> 5;
    for (int t = wave; t < NTILE * NTILE; t += nwaves) {
        const int ti = t / NTILE, tj = t % NTILE;
        v2f av, bv;
        // A (16x4, MxK): lanes 0-15 hold K=0,1; lanes 16-31 hold K=2,3 (zero)
        av[0] = half ? 0.f : sarea[ti * 16 + l16];   // K=0 col = area_i
        av[1] = half ? 0.f : 1.0f;                   // K=1 col = 1
        // B (4x16, KxN): lanes 0-15 hold rows K=0,1; lanes 16-31 K=2,3 (zero)
        bv[0] = half ? 0.f : 1.0f;                   // K=0 row = 1
        bv[1] = half ? 0.f : sarea[tj * 16 + l16];   // K=1 row = area_j
        v8f cz = {0.f, 0.f, 0.f, 0.f, 0.f, 0.f, 0.f, 0.f};
        // D[i][j] = area_i + area_j   (exact in f32)
        v8f d = __builtin_amdgcn_wmma_f32_16x16x4_f32(
            false, av, false, bv, (short)0, cz, false, false);

        const int j    = tj * 16 + l16;              // D layout: N = lane & 15
        const int jw   = tj >> 1;                    // 16-bit group -> word
        const int jsh  = (tj & 1) * 16;              // shift within word
        const float x2j = sx2[j], x1j = sx1[j], y2j = sy2[j], y1j = sy1[j];
#pragma unroll
        for (int v = 0; v < 8; ++v) {
            const int i = ti * 16 + v + half * 8;    // D layout: M = v (+8 hi half)
            float ix = fminf(sx2[i], x2j) - fmaxf(sx1[i], x1j);
            float iy = fminf(sy2[i], y2j) - fmaxf(sy1[i], y1j);
            float inter = fmaxf(ix, 0.f) * fmaxf(iy, 0.f);
            float uni = d[v] - inter;                // area_i + area_j - inter
            // inter/max(uni,eps) > T  <=>  inter > T*max(uni,eps)
            bool sup = (inter > IOU_T * fmaxf(uni, 1e-8f)) && (j > i);
            unsigned long long bits = __ballot(sup);
            if (lane == 0) {
                unsigned mLo = (unsigned)(bits & 0xFFFFu);         // row ti*16+v
                unsigned mHi = (unsigned)((bits >> 16) & 0xFFFFu); // row +8
                if (mLo) atomicOr(&rows[ti * 16 + v][jw],     mLo << jsh);
                if (mHi) atomicOr(&rows[ti * 16 + v + 8][jw], mHi << jsh);
            }
        }
    }
    __syncthreads();

    // exact greedy sweep (matches jax fori_loop semantics)
    if (tid == 0) {
        unsigned kw[ROWW];
#pragma unroll
        for (int w = 0; w < ROWW; ++w) kw[w] = 0u;
        for (int k = 0; k < K_PER; ++k)
            if (sscore[k] > CONF_T) kw[k >> 5] |= 1u << (k & 31);
        for (int i = 0; i < K_PER; ++i) {
            if (kw[i >> 5] & (1u << (i & 31))) {
#pragma unroll
                for (int w = 0; w < ROWW; ++w) kw[w] &= ~rows[i][w];
            }
        }
#pragma unroll
        for (int w = 0; w < ROWW; ++w) keepw[w] = kw[w];
    }
    __syncthreads();

    for (int k = tid; k < K_PER; k += blockDim.x) {
        bool kept = (keepw[k >> 5] >> (k & 31)) & 1u;
        ws_scores[(size_t)bc * K_PER + k] = kept ? sscore[k] : -1.0f;
    }
}

// ---------------------------------------------------------------------------
// Kernel 3: per-batch combined top-200 over C*200 kept scores.
// Same bucket-select + bitonic machinery; writes final outputs.
// out layout (flat): boxes[B*200*4] | scores[B*200] | classes[B*200] | valid[B]
// ---------------------------------------------------------------------------
__global__ void __launch_bounds__(256)
combine_topk(const float* __restrict__ ws_scores,
             const float* __restrict__ ws_boxes,
             int C, int B,
             float* __restrict__ out) {
    const int b   = blockIdx.x;
    const int tid = threadIdx.x;
    const int N   = C * K_PER;

    __shared__ unsigned hist[NBUCKET];
    __shared__ unsigned candKey[CAND_MAX];
    __shared__ int      candIdx[CAND_MAX];
    __shared__ int      candCnt;
    __shared__ unsigned pivotKey;

    for (int i = tid; i < NBUCKET; i += blockDim.x) hist[i] = 0u;
    if (tid == 0) candCnt = 0;
    __syncthreads();

    const float* sc = ws_scores + (size_t)b * N;
    for (int f = tid; f < N; f += blockDim.x)
        atomicAdd(&hist[flipKey(sc[f]) >> 21], 1u);
    __syncthreads();

    if (tid == 0) {
        unsigned cum = 0; int p = 0;
        for (int i = NBUCKET - 1; i >= 0; --i) {
            cum += hist[i];
            if (cum >= MAX_DET) { p = i; break; }
        }
        pivotKey = ((unsigned)p) << 21;
    }
    __syncthreads();
    const unsigned pk = pivotKey;

    for (int f = tid; f < N; f += blockDim.x) {
        float s = sc[f];
        unsigned key = flipKey(s);
        if (s > 0.f && key >= pk) {         // only real detections can be "ok"
            int pos = atomicAdd(&candCnt, 1);
            if (pos < CAND_MAX) { candKey[pos] = ~key; candIdx[pos] = f; }
        }
    }
    __syncthreads();
    int cnt = candCnt; if (cnt > CAND_MAX) cnt = CAND_MAX;
    for (int i = cnt + tid; i < CAND_MAX; i += blockDim.x) {
        candKey[i] = 0xFFFFFFFFu; candIdx[i] = 0;
    }
    __syncthreads();

    bitonic_sort_pairs(candKey, candIdx, tid, blockDim.x);

    const int valid = cnt < MAX_DET ? cnt : MAX_DET;
    float* outBoxes   = out;
    float* outScores  = out + (size_t)B * MAX_DET * 4;
    float* outClasses = out + (size_t)B * MAX_DET * 5;
    float* outValid   = out + (size_t)B * MAX_DET * 6;

    for (int k = tid; k < MAX_DET; k += blockDim.x) {
        float s = 0.f, cl = 0.f, bx0 = 0.f, bx1 = 0.f, bx2 = 0.f, bx3 = 0.f;
        if (k < valid) {
            s = unflipKey(~candKey[k]);
            int f = candIdx[k];
            cl = (float)(f / K_PER);
            const float* pb = ws_boxes + ((size_t)b * N + f) * 4;
            bx0 = pb[0]; bx1 = pb[1]; bx2 = pb[2]; bx3 = pb[3];
        }
        float* ob = outBoxes + ((size_t)b * MAX_DET + k) * 4;
        ob[0] = bx0; ob[1] = bx1; ob[2] = bx2; ob[3] = bx3;
        outScores [(size_t)b * MAX_DET + k] = s;
        outClasses[(size_t)b * MAX_DET + k] = cl;
    }
    if (tid == 0) outValid[b] = (float)valid;
}

// ---------------------------------------------------------------------------
extern "C" void kernel_launch(void* const* d_in, const int* in_sizes, int n_in,
                              void* d_out, int out_size, void* d_ws, size_t ws_size,
                              hipStream_t stream) {
    const float* logits  = (const float*)d_in[0];  // [B, A, C]
    const float* reg     = (const float*)d_in[1];  // [B, A, 4]
    const float* anchors = (const float*)d_in[2];  // [A, 4]

    const int A = in_sizes[2] / 4;
    const int B = in_sizes[1] / (A * 4);
    const int C = in_sizes[0] / (A * B);

    // workspace partition (~3.9 MB for B=8, C=80)
    float* ws_vals   = (float*)d_ws;                               // B*C*256
    int*   ws_idx    = (int*)(ws_vals + (size_t)B * C * 256);      // B*C*256
    float* ws_scores = (float*)(ws_idx + (size_t)B * C * 256);     // B*C*200
    float* ws_boxes  = ws_scores + (size_t)B * C * K_PER;          // B*C*200*4

    topk_per_class<<<B * C, 256, 0, stream>>>(logits, A, C, ws_vals, ws_idx);
    nms_per_class <<<B * C, 256, 0, stream>>>(reg, anchors, ws_vals, ws_idx,
                                              A, C, ws_scores, ws_boxes);
    combine_topk  <<<B, 256, 0, stream>>>(ws_scores, ws_boxes, C, B,
                                          (float*)d_out);
}